// ButterWorthLayer_19189913878721
// MI455X (gfx1250) — compile-verified
//
#include <hip/hip_runtime.h>
#include <cstdint>
#include <cstddef>

#ifndef __has_builtin
#define __has_builtin(x) 0
#endif

// ---------------- problem constants ----------------
#define NCHAN   2048        // 32*64 channels
#define TLEN    16384       // samples per channel
#define LBLK    1024        // time-block length
#define NBLK    (TLEN/LBLK) // 16 blocks per channel
#define NTASK   (NCHAN*NBLK)// 32768 (channel,block) tasks
#define NS      6           // SOS sections (5th order HP -> 3, LP -> 3)
#define NST     12          // state dim = 2*NS
#define CH      32          // samples per LDS chunk
#define NCHUNK  (LBLK/CH)   // 32 chunks per block
#define PADW    36          // padded words per thread chunk (16B aligned, bank-spread)
#define TPB_A   128
#define NBUF    3           // triple-buffered async pipeline (2 chunks in flight)

typedef __attribute__((ext_vector_type(2))) float v2f;
typedef __attribute__((ext_vector_type(8))) float v8f;
typedef __attribute__((ext_vector_type(4))) int   v4i;

// ---------------- CDNA5 async-copy helpers ----------------
#if __has_builtin(__builtin_amdgcn_s_wait_asynccnt)
#define WAIT_ASYNCCNT(n) __builtin_amdgcn_s_wait_asynccnt(n)
#else
#define WAIT_ASYNCCNT(n) asm volatile("s_wait_asynccnt %0" :: "i"(n) : "memory")
#endif

__device__ __forceinline__ void async_ld_b128(const float* gsrc, float* ldst) {
#if __has_builtin(__builtin_amdgcn_global_load_async_to_lds_b128)
    __builtin_amdgcn_global_load_async_to_lds_b128(
        (__attribute__((address_space(1))) v4i*)(uintptr_t)gsrc,
        (__attribute__((address_space(3))) v4i*)(uint32_t)(uintptr_t)ldst,
        0, 0);
#else
    asm volatile("global_load_async_to_lds_b128 %0, %1, off"
                 :: "v"((uint32_t)(uintptr_t)ldst),
                    "v"((uint64_t)(uintptr_t)gsrc)
                 : "memory");
#endif
}

// =====================================================================
// Kernel 1: per-(channel,block) cascade with zero initial state.
// Writes partial output y0 and driven final state v (12 floats).
// Input staged through LDS with triple-buffered async copies.
// =====================================================================
__global__ __launch_bounds__(TPB_A) void k_blocks(
    const float* __restrict__ x, const float* __restrict__ sos,
    float* __restrict__ y0, float* __restrict__ vst)
{
    __shared__ float buf[NBUF][TPB_A * PADW];
    const int task = blockIdx.x * TPB_A + threadIdx.x;   // grid is exact
    const float* xp = x + (size_t)task * LBLK;           // (c*NBLK+b)*LBLK == c*TLEN+b*LBLK

    float b0[NS], b1c[NS], b2c[NS], a1c[NS], a2c[NS];
#pragma unroll
    for (int i = 0; i < NS; ++i) {
        b0[i]  = sos[6*i + 0];
        b1c[i] = sos[6*i + 1];
        b2c[i] = sos[6*i + 2];
        a1c[i] = sos[6*i + 4];
        a2c[i] = sos[6*i + 5];
    }
    float w1[NS], w2[NS];
#pragma unroll
    for (int i = 0; i < NS; ++i) { w1[i] = 0.f; w2[i] = 0.f; }

    // prefetch chunks 0 and 1
#pragma unroll
    for (int j = 0; j < CH/4; ++j)
        async_ld_b128(xp + 4*j, &buf[0][threadIdx.x * PADW] + 4*j);
#pragma unroll
    for (int j = 0; j < CH/4; ++j)
        async_ld_b128(xp + CH + 4*j, &buf[1][threadIdx.x * PADW] + 4*j);

    for (int chk = 0; chk < NCHUNK; ++chk) {
        const float* cur = &buf[chk % NBUF][threadIdx.x * PADW];
        if (chk + 2 < NCHUNK) {
            float* nxt = &buf[(chk + 2) % NBUF][threadIdx.x * PADW];
            const float* src = xp + (chk + 2) * CH;
#pragma unroll
            for (int j = 0; j < CH/4; ++j) async_ld_b128(src + 4*j, nxt + 4*j);
            WAIT_ASYNCCNT(16);     // 2 newest chunks may remain outstanding
        } else if (chk + 1 < NCHUNK) {
            WAIT_ASYNCCNT(8);      // 1 newest chunk may remain outstanding
        } else {
            WAIT_ASYNCCNT(0);
        }

        float o4[4];
#pragma unroll
        for (int t = 0; t < CH; ++t) {
            float u = cur[t];
#pragma unroll
            for (int i = 0; i < NS; ++i) {
                // DF2T biquad: y = b0*u + w1; w1' = b1*u - a1*y + w2; w2' = b2*u - a2*y
                float y = fmaf(b0[i], u, w1[i]);
                w1[i] = fmaf(b1c[i], u, w2[i]) - a1c[i] * y;
                w2[i] = fmaf(b2c[i], u, -(a2c[i] * y));
                u = y;
            }
            o4[t & 3] = u;
            if ((t & 3) == 3) {
                float4 st = make_float4(o4[0], o4[1], o4[2], o4[3]);
                *(float4*)(y0 + (size_t)task * LBLK + chk * CH + (t - 3)) = st;
            }
        }
    }
    // state row is 48B -> 16B aligned; 3 x b128 stores
    float* vp = vst + (size_t)task * NST;
    *(float4*)(vp + 0) = make_float4(w1[0], w2[0], w1[1], w2[1]);
    *(float4*)(vp + 4) = make_float4(w1[2], w2[2], w1[3], w2[3]);
    *(float4*)(vp + 8) = make_float4(w1[4], w2[4], w1[5], w2[5]);
}

// =====================================================================
// Kernel 2: build constants (double precision):
//   F : 12x12 one-step state map of the cascade,  c : output row
//   W[t] = c^T F^t  (t = 0..LBLK-1),  M = F^LBLK (by repeated squaring,
//   each squaring parallel over 144 threads)
// =====================================================================
__global__ __launch_bounds__(1024) void k_prep(
    const float* __restrict__ sos, float* __restrict__ W, float* __restrict__ M)
{
    __shared__ double P[11][NST][NST];   // P[k] = F^(2^k)
    __shared__ double cvecS[NST];

    if (threadIdx.x == 0) {
        // Symbolic single-step: each value is affine over (state[12], x).
        double u[NST + 1];
        for (int k = 0; k <= NST; ++k) u[k] = (k == NST) ? 1.0 : 0.0;
        double Fg[NST][NST + 1];
        for (int i = 0; i < NS; ++i) {
            double bb0 = sos[6*i+0], bb1 = sos[6*i+1], bb2 = sos[6*i+2];
            double aa1 = sos[6*i+4], aa2 = sos[6*i+5];
            double y[NST + 1];
            for (int k = 0; k <= NST; ++k) y[k] = bb0 * u[k];
            y[2*i + 0] += 1.0;                                  // + w1_i
            for (int k = 0; k <= NST; ++k) Fg[2*i+0][k] = bb1 * u[k] - aa1 * y[k];
            Fg[2*i+0][2*i + 1] += 1.0;                          // + w2_i
            for (int k = 0; k <= NST; ++k) Fg[2*i+1][k] = bb2 * u[k] - aa2 * y[k];
            for (int k = 0; k <= NST; ++k) u[k] = y[k];
        }
        for (int r = 0; r < NST; ++r)
            for (int cc = 0; cc < NST; ++cc)
                P[0][r][cc] = Fg[r][cc];
        for (int k = 0; k < NST; ++k) cvecS[k] = u[k];
    }
    __syncthreads();

    // parallel repeated squaring: P[k] = P[k-1]^2, one element per thread
    for (int k = 1; k <= 10; ++k) {
        if (threadIdx.x < NST * NST) {
            const int r = threadIdx.x / NST, cc = threadIdx.x % NST;
            double acc = 0.0;
#pragma unroll
            for (int m = 0; m < NST; ++m) acc += P[k-1][r][m] * P[k-1][m][cc];
            P[k][r][cc] = acc;
        }
        __syncthreads();
    }

    const int t = threadIdx.x;            // 0..1023
    double r[NST];
    for (int j = 0; j < NST; ++j) r[j] = cvecS[j];
    for (int k = 0; k < 10; ++k) {
        if ((t >> k) & 1) {
            double rn[NST];
            for (int j = 0; j < NST; ++j) {
                double acc = 0.0;
                for (int m = 0; m < NST; ++m) acc += r[m] * P[k][m][j];
                rn[j] = acc;
            }
            for (int j = 0; j < NST; ++j) r[j] = rn[j];
        }
    }
    for (int j = 0; j < NST; ++j) W[(size_t)t * NST + j] = (float)r[j];
    if (t < NST * NST) M[t] = (float)P[10][t / NST][t % NST];
}

// =====================================================================
// Kernel 3: per-channel block scan  s_in[b+1] = M*s_in[b] + v[b]
// =====================================================================
__global__ __launch_bounds__(256) void k_scan(
    const float* __restrict__ vst, const float* __restrict__ M,
    float* __restrict__ sinb)
{
    __shared__ float Ms[NST * NST];
    if (threadIdx.x < NST * NST) Ms[threadIdx.x] = M[threadIdx.x];
    __syncthreads();
    const int c = blockIdx.x * 256 + threadIdx.x;   // 0..NCHAN-1 (grid exact)
    float s[NST];
#pragma unroll
    for (int j = 0; j < NST; ++j) s[j] = 0.f;
    for (int b = 0; b < NBLK; ++b) {
        const size_t task = (size_t)c * NBLK + b;
#pragma unroll
        for (int j = 0; j < NST; ++j) sinb[task * NST + j] = s[j];
        float ns[NST];
#pragma unroll
        for (int rr = 0; rr < NST; ++rr) {
            float acc = vst[task * NST + rr];
#pragma unroll
            for (int m = 0; m < NST; ++m) acc = fmaf(Ms[rr * NST + m], s[m], acc);
            ns[rr] = acc;
        }
#pragma unroll
        for (int j = 0; j < NST; ++j) s[j] = ns[j];
    }
}

// =====================================================================
// Kernel 4: exact correction  y[task][t] += W[t] . s_in[task]
// Batched as 16(t) x 16(task) x 12(k) f32 matmuls -> V_WMMA_F32_16X16X4_F32.
// One wave per 16x16 output tile; 3 accumulating K=4 WMMAs.
// D layout (32-bit 16x16): VGPR r, lane l: Mrow = r + 8*(l>=16), Ncol = l&15.
// =====================================================================
__global__ __launch_bounds__(256) void k_corr(
    float* __restrict__ y, const float* __restrict__ W,
    const float* __restrict__ sinb)
{
    const int wave  = blockIdx.x * 8 + (threadIdx.x >> 5);
    const int lane  = threadIdx.x & 31;
    const int nTile = wave & (NTASK / 16 - 1);   // 2048 task tiles
    const int tTile = wave >> 11;                // 64 time tiles
    const int n0 = nTile * 16, t0 = tTile * 16;
    const int lm = lane & 15, half = lane >> 4;

    v8f acc = {0.f, 0.f, 0.f, 0.f, 0.f, 0.f, 0.f, 0.f};
#pragma unroll
    for (int k0 = 0; k0 < NST; k0 += 4) {
        // A(16x4) = W[t0+m][k0+k]; lanes 0-15 hold k={0,1}, lanes 16-31 k={2,3}
        const float* wr = W + (size_t)(t0 + lm) * NST + k0 + 2 * half;
        v2f a; a.x = wr[0]; a.y = wr[1];
        // B(4x16) = S[k0+k][n0+n]; mirrored K split across lane halves
        const float* sr = sinb + (size_t)(n0 + lm) * NST + k0 + 2 * half;
        v2f b; b.x = sr[0]; b.y = sr[1];
        acc = __builtin_amdgcn_wmma_f32_16x16x4_f32(
            false, a, false, b, (short)0, acc, false, false);
    }

    // lane holds 8 consecutive time samples of task n0+lm
    const size_t base = (size_t)(n0 + lm) * LBLK + t0 + 8 * half;
    float4 ya = *(const float4*)(y + base);
    float4 yb = *(const float4*)(y + base + 4);
    ya.x += acc[0]; ya.y += acc[1]; ya.z += acc[2]; ya.w += acc[3];
    yb.x += acc[4]; yb.y += acc[5]; yb.z += acc[6]; yb.w += acc[7];
    *(float4*)(y + base)     = ya;
    *(float4*)(y + base + 4) = yb;
}

// =====================================================================
extern "C" void kernel_launch(void* const* d_in, const int* in_sizes, int n_in,
                              void* d_out, int out_size, void* d_ws, size_t ws_size,
                              hipStream_t stream)
{
    const float* x   = (const float*)d_in[0];   // (32,64,16384) f32
    const float* sos = (const float*)d_in[1];   // (6,6) f32
    float* yout = (float*)d_out;
    float* ws   = (float*)d_ws;

    float* vst  = ws;                            // NTASK*12  = 393216 f
    float* sinb = ws + (size_t)NTASK * NST;      // NTASK*12  = 393216 f
    float* W    = sinb + (size_t)NTASK * NST;    // LBLK*12   = 12288 f
    float* M    = W + (size_t)LBLK * NST;        // 144 f

    k_blocks<<<NTASK / TPB_A, TPB_A, 0, stream>>>(x, sos, yout, vst);
    k_prep  <<<1, 1024, 0, stream>>>(sos, W, M);
    k_scan  <<<NCHAN / 256, 256, 0, stream>>>(vst, M, sinb);
    k_corr  <<<(NTASK / 16) * (LBLK / 16) / 8, 256, 0, stream>>>(yout, W, sinb);
}